// SSDLoss_44590350467175
// MI455X (gfx1250) — compile-verified
//
#include <hip/hip_runtime.h>
#include <stdint.h>

// ---------------------------------------------------------------------------
// SSD hard-negative-mining loss for MI455X (gfx1250, wave32, WGP, 320KB LDS)
//   predicts: (64, 2, 512, 512) f32 logits
//   targets : (64, 1, 512, 512) int  (0/1)
//   out     : scalar f32
// One workgroup per batch row. Exact top-k negative sum via 2-level (16+16 bit)
// radix select on the float bit pattern (con >= 0 -> bits are order-preserving).
// CE math uses hardware v_exp_f32/v_log_f32 (TRANS pipe) to stay on the
// 23.3 TB/s memory roofline instead of the precise log1p VALU expansion.
// ---------------------------------------------------------------------------

#define BATCH      64
#define NPIX       (512 * 512)            // 262144 pixels per batch
#define THREADS    1024                   // 32 wave32 waves
#define VEC        4                      // float4 / int4 per thread per iter
#define ELEMS_ITER (THREADS * VEC)        // 4096 elements per block iteration
#define ITERS      (NPIX / ELEMS_ITER)    // 64 iterations
#define NBINS      65536                  // 16-bit radix level
#define BINS_PER_T (NBINS / THREADS)      // 64
#define NEGF       3u

typedef __attribute__((ext_vector_type(2))) float v2f;
typedef __attribute__((ext_vector_type(8))) float v8f;

// ---- dynamic LDS layout (32-bit words) ------------------------------------
#define HIST_OFF   0                          // 65536 words (256 KB) histogram
#define FSC_OFF    (HIST_OFF + NBINS)         // 32 words float reduce scratch
#define WSC_OFF    (FSC_OFF + 32)             // 32 words wave-suffix scratch
#define ICTL_OFF   (WSC_OFF + 32)             // 8 words control
#define STAGE_OFF  (ICTL_OFF + 8)             // 16B aligned (65608*4 % 16 == 0)
#define STAGE_INTS (2 * ELEMS_ITER)           // 8192 words: double-buffer targets
#define SMEM_WORDS (STAGE_OFF + STAGE_INTS)
#define SMEM_BYTES ((size_t)SMEM_WORDS * 4)   // 295200 B <= 320 KB per WGP

// ---- wave32 helpers --------------------------------------------------------
__device__ __forceinline__ float wave_sum32(float v) {
#pragma unroll
  for (int off = 16; off; off >>= 1) v += __shfl_xor(v, off, 32);
  return v;
}
__device__ __forceinline__ unsigned wave_sum32u(unsigned v) {
#pragma unroll
  for (int off = 16; off; off >>= 1) v += __shfl_xor(v, off, 32);
  return v;
}

// block-wide sum over 1024 threads (32 waves), result broadcast to all threads
__device__ __forceinline__ float block_sum1024(float v, float* fsc, int t) {
  v = wave_sum32(v);
  const int w = t >> 5, lane = t & 31;
  if (lane == 0) fsc[w] = v;
  __syncthreads();
  if (t < 32) {
    float r = fsc[t];
    r = wave_sum32(r);
    if (t == 0) fsc[0] = r;
  }
  __syncthreads();
  float r = fsc[0];
  __syncthreads();
  return r;
}

// ---- 2-class cross entropy -------------------------------------------------
// con = lse(l0,l1) - l_tgt = softplus(l_other - l_tgt).
// With e = l1 - l0:  |l_other - l_tgt| == |e| for either target, so one
// subtract + one exp + one log (hardware TRANS ops) covers both cases.
__device__ __forceinline__ float ce2(float l0, float l1, int tgt) {
  const float e    = l1 - l0;
  const float tail = __logf(1.0f + __expf(-fabsf(e)));  // v_exp_f32 + v_log_f32
  const float dpos = (tgt != 0) ? -e : e;               // l_other - l_tgt
  return fmaxf(dpos, 0.0f) + tail;
}

// ---- matrix-pipe reduction: ones(16x4) x B(4x16) accumulated into C --------
// C[m][n] += sum_k B[k][n]; sum over all lanes of C row 0 == 2 * total of the
// 64 f32 values fed in B (2 per lane). Offloads the streaming adds to the
// tensor pipe (v_wmma_f32_16x16x4_f32), leaving VALU/TRANS free for CE math.
__device__ __forceinline__ v8f wmma_sum_pair(float x, float y, v8f acc) {
  v2f a; a.x = 1.0f; a.y = 1.0f;
  v2f b; b.x = x;    b.y = y;
  return __builtin_amdgcn_wmma_f32_16x16x4_f32(false, a, false, b,
                                               (short)0, acc, false, false);
}
__device__ __forceinline__ float wmma_acc_total(v8f acc) {
  return wave_sum32(acc[0]) * 0.5f;  // row 0 appears twice across 32 lanes
}

// ---- radix select step: find bin T and residual r with suffix count >= k ---
__device__ __forceinline__ void radix_scan(unsigned* hist, unsigned* wsc,
                                           unsigned* outT, unsigned* outR,
                                           unsigned k, int t) {
  const int lane = t & 31, w = t >> 5;
  const int base = t * BINS_PER_T;
  unsigned s = 0;
  for (int j = 0; j < BINS_PER_T; ++j) s += hist[base + j];
  // inclusive suffix-sum of per-thread chunk counts within the wave
  unsigned inc = s;
#pragma unroll
  for (int off = 1; off < 32; off <<= 1) {
    unsigned u = __shfl_down(inc, off, 32);
    if (lane + off < 32) inc += u;
  }
  if (lane == 0) wsc[w] = inc;  // wave totals
  __syncthreads();
  if (t < 32) {                 // suffix over the 32 wave totals (wave 0)
    unsigned wv = wsc[t];
    unsigned winc = wv;
#pragma unroll
    for (int off = 1; off < 32; off <<= 1) {
      unsigned u = __shfl_down(winc, off, 32);
      if (t + off < 32) winc += u;
    }
    wsc[t] = winc - wv;         // counts in waves strictly after this one
  }
  __syncthreads();
  const unsigned after = (inc - s) + wsc[w];  // counts in bins above my chunk
  if (after < k && k <= after + s) {          // exactly one thread matches
    unsigned cum = after;
    for (int j = BINS_PER_T - 1; j >= 0; --j) {
      const unsigned h = hist[base + j];
      if (cum + h >= k) { *outT = (unsigned)(base + j); *outR = k - cum; break; }
      cum += h;
    }
  }
  __syncthreads();
}

// ---------------------------------------------------------------------------
extern "C" __global__ void __launch_bounds__(THREADS, 1)
ssd_hnm_batch(const float* __restrict__ predicts,
              const int*  __restrict__ targets,
              float* __restrict__ batch_loss) {
  extern __shared__ unsigned int smem[];
  unsigned int* hist  = smem + HIST_OFF;
  float*        fsc   = (float*)(smem + FSC_OFF);
  unsigned int* wsc   = smem + WSC_OFF;
  unsigned int* ictl  = smem + ICTL_OFF;
  int*          stage = (int*)(smem + STAGE_OFF);

  const int b = blockIdx.x;
  const int t = threadIdx.x;
  const float* p0 = predicts + (size_t)b * (2 * NPIX);
  const float* p1 = p0 + NPIX;
  const int*   tg = targets + (size_t)b * NPIX;

  {  // vector clear of the histogram: 16 x ds_store_b128 per thread
    const uint4 z = {0u, 0u, 0u, 0u};
    uint4* h4 = (uint4*)hist;
    for (int i = t; i < NBINS / 4; i += THREADS) h4[i] = z;
  }
  if (t < 8) ictl[t] = 0u;
  __syncthreads();

  // ================= Phase 1: stream HBM, stats + coarse histogram =========
  v8f accAll = {0.f, 0.f, 0.f, 0.f, 0.f, 0.f, 0.f, 0.f};
  float   sumPos = 0.0f;
  unsigned posCnt = 0u;

  // prologue: async-stage targets tile 0 into LDS (CDNA5 async copy path)
  {
    unsigned lds = (unsigned)(uintptr_t)(stage + t * VEC);
    unsigned long long g = (unsigned long long)(uintptr_t)(tg + t * VEC);
    asm volatile("global_load_async_to_lds_b128 %0, %1, off"
                 :: "v"(lds), "v"(g) : "memory");
  }
  for (int it = 0; it < ITERS; ++it) {
    const int buf = it & 1;
    if (it + 1 < ITERS) {
      unsigned lds = (unsigned)(uintptr_t)(stage + (buf ^ 1) * ELEMS_ITER + t * VEC);
      unsigned long long g =
          (unsigned long long)(uintptr_t)(tg + (it + 1) * ELEMS_ITER + t * VEC);
      asm volatile("global_load_async_to_lds_b128 %0, %1, off"
                   :: "v"(lds), "v"(g) : "memory");
      asm volatile("s_wait_asynccnt 1" ::: "memory");
    } else {
      asm volatile("s_wait_asynccnt 0" ::: "memory");
    }
    const int base = it * ELEMS_ITER + t * VEC;
    const float4 a0 = *(const float4*)(p0 + base);
    const float4 a1 = *(const float4*)(p1 + base);
    const int4   tv = *(const int4*)(stage + buf * ELEMS_ITER + t * VEC);

    const float c0 = ce2(a0.x, a1.x, tv.x);
    const float c1 = ce2(a0.y, a1.y, tv.y);
    const float c2 = ce2(a0.z, a1.z, tv.z);
    const float c3 = ce2(a0.w, a1.w, tv.w);

    // matrix-pipe accumulation of sum(con) -- uniform control flow, EXEC all 1
    accAll = wmma_sum_pair(c0, c1, accAll);
    accAll = wmma_sum_pair(c2, c3, accAll);

    const int   tt[4] = {tv.x, tv.y, tv.z, tv.w};
    const float cc[4] = {c0, c1, c2, c3};
#pragma unroll
    for (int j = 0; j < 4; ++j) {
      const bool pos = tt[j] > 0;
      sumPos += pos ? cc[j] : 0.0f;
      posCnt += pos ? 1u : 0u;
      if (!pos) atomicAdd(&hist[__float_as_uint(cc[j]) >> 16], 1u);  // ds_add
    }
  }

  const float sumAll =
      block_sum1024(((t & 31) == 0) ? wmma_acc_total(accAll) : 0.0f, fsc, t);
  const float sumPosT = block_sum1024(sumPos, fsc, t);
  const unsigned wPos = wave_sum32u(posCnt);
  if ((t & 31) == 0) atomicAdd(&ictl[4], wPos);
  __syncthreads();

  const unsigned posT = ictl[4];
  const unsigned negT = (unsigned)NPIX - posT;
  unsigned k = NEGF * posT;
  if (k > (unsigned)NPIX) k = (unsigned)NPIX;

  float closs = 0.0f;
  const bool doSel = (posT > 0u) && (k < negT);   // block-uniform
  if (posT > 0u && !doSel) closs = sumAll;        // all negatives selected

  if (doSel) {
    // ============ Phase 2: coarse (bits[31:16]) threshold ==================
    radix_scan(hist, wsc, &ictl[0], &ictl[1], k, t);
    const unsigned T = ictl[0];
    const unsigned r = ictl[1];
    {  // clear histogram again for the fine level
      const uint4 z = {0u, 0u, 0u, 0u};
      uint4* h4 = (uint4*)hist;
      for (int i = t; i < NBINS / 4; i += THREADS) h4[i] = z;
    }
    __syncthreads();

    // ============ Phase 3: re-stream (L2-resident, 3MB/batch) ==============
    v8f accSel = {0.f, 0.f, 0.f, 0.f, 0.f, 0.f, 0.f, 0.f};
    for (int it = 0; it < ITERS; ++it) {
      const int base = it * ELEMS_ITER + t * VEC;
      const float4 a0 = *(const float4*)(p0 + base);
      const float4 a1 = *(const float4*)(p1 + base);
      const int4   tv = *(const int4*)(tg + base);
      const int   tt[4] = {tv.x, tv.y, tv.z, tv.w};
      float cc[4];
      cc[0] = ce2(a0.x, a1.x, tv.x);
      cc[1] = ce2(a0.y, a1.y, tv.y);
      cc[2] = ce2(a0.z, a1.z, tv.z);
      cc[3] = ce2(a0.w, a1.w, tv.w);
      float sel[4];
      unsigned bits[4];
#pragma unroll
      for (int j = 0; j < 4; ++j) {
        bits[j] = __float_as_uint(cc[j]);
        const bool isNeg = (tt[j] <= 0);
        sel[j] = (isNeg && (bits[j] >> 16) > T) ? cc[j] : 0.0f;
      }
      accSel = wmma_sum_pair(sel[0], sel[1], accSel);
      accSel = wmma_sum_pair(sel[2], sel[3], accSel);
#pragma unroll
      for (int j = 0; j < 4; ++j)
        if (tt[j] <= 0 && (bits[j] >> 16) == T)
          atomicAdd(&hist[bits[j] & 0xFFFFu], 1u);
    }
    __syncthreads();

    // ============ Phase 4: fine (bits[15:0]) — bins are exact f32 values ===
    radix_scan(hist, wsc, &ictl[2], &ictl[3], r, t);
    const unsigned T2 = ictl[2];
    const unsigned r2 = ictl[3];
    float fs = 0.0f;
    const int bb = t * BINS_PER_T;
    for (int j = 0; j < BINS_PER_T; ++j) {
      const unsigned bin = (unsigned)(bb + j);
      if (bin > T2)
        fs += (float)hist[bin] * __uint_as_float((T << 16) | bin);
    }
    if (t == 0) fs += (float)r2 * __uint_as_float((T << 16) | T2);
    const float fineSum = block_sum1024(fs, fsc, t);
    const float selSum =
        block_sum1024(((t & 31) == 0) ? wmma_acc_total(accSel) : 0.0f, fsc, t);
    closs = sumPosT + selSum + fineSum;
  }

  if (t == 0) {
    float loss = 0.0f;
    if (posT > 0u) loss = closs / fmaxf((float)posT, 1e-6f);
    batch_loss[b] = loss;
  }
}

// ---- final: mean over 64 batch losses -------------------------------------
extern "C" __global__ void ssd_hnm_final(const float* __restrict__ batch_loss,
                                         float* __restrict__ out) {
  const int t = threadIdx.x;  // 64 threads = 2 waves
  float v = batch_loss[t];
  v = wave_sum32(v);
  __shared__ float w[2];
  if ((t & 31) == 0) w[t >> 5] = v;
  __syncthreads();
  if (t == 0) out[0] = (w[0] + w[1]) * (1.0f / (float)BATCH);
}

// ---------------------------------------------------------------------------
extern "C" void kernel_launch(void* const* d_in, const int* in_sizes, int n_in,
                              void* d_out, int out_size, void* d_ws, size_t ws_size,
                              hipStream_t stream) {
  const float* predicts = (const float*)d_in[0];
  const int*   targets  = (const int*)d_in[1];
  float* batch_loss = (float*)d_ws;  // 64 floats of scratch

  ssd_hnm_batch<<<dim3(BATCH), dim3(THREADS), SMEM_BYTES, stream>>>(
      predicts, targets, batch_loss);
  ssd_hnm_final<<<dim3(1), dim3(BATCH), 0, stream>>>(batch_loss, (float*)d_out);

  (void)in_sizes; (void)n_in; (void)out_size; (void)ws_size;
}